// RipEncoding_55095840473138
// MI455X (gfx1250) — compile-verified
//
#include <hip/hip_runtime.h>
#include <math.h>

#define NS 262144      // samples
#define NV 10          // vertices
#define FEAT 16
#define RES 512
#define T_PIX 1040400  // flat ripmap pixels per vertex (sum over 8x8 planes)

typedef __attribute__((ext_vector_type(2))) float v2f;
typedef __attribute__((ext_vector_type(8))) float v8f;

__host__ __device__ inline int plane_off(int l1, int l2) {
    // cum width sum: sum_{a<l} (512>>a) = 1024 - (1024>>l); sum_l H[l] = 1020
    return 1020 * (1024 - (1024 >> l1)) + (512 >> l1) * (1024 - (1024 >> l2));
}

// ---------------- Setup: build 12x48 coefficient matrix B for the projection GEMM ----------------

__device__ __constant__ double c_verts[10][3] = {
    {1.0, 1.0, 1.0}, {1.0, 1.0, -1.0}, {1.0, -1.0, 1.0}, {1.0, -1.0, -1.0},
    {0.0, 1.6180339887498948482, 0.61803398874989484820},
    {0.0, 1.6180339887498948482, -0.61803398874989484820},
    {0.61803398874989484820, 0.0, 1.6180339887498948482},
    {0.61803398874989484820, 0.0, -1.6180339887498948482},
    {1.6180339887498948482, 0.61803398874989484820, 0.0},
    {1.6180339887498948482, -0.61803398874989484820, 0.0}};

__device__ void proj_axes(int v, float p0f[3], float p1f[3]) {
    double a0 = c_verts[v][0], a1 = c_verts[v][1], a2 = c_verts[v][2];
    double n = sqrt(a0 * a0 + a1 * a1 + a2 * a2);
    a0 /= n; a1 /= n; a2 /= n;
    // all verts have a0!=0 || a1!=0
    double p0x = -a1, p0y = a0, p0z = 0.0;
    double p1x = a1 * p0z - a2 * p0y;
    double p1y = a2 * p0x - a0 * p0z;
    double p1z = a0 * p0y - a1 * p0x;
    double n0 = sqrt(p0x * p0x + p0y * p0y + p0z * p0z);
    double n1 = sqrt(p1x * p1x + p1y * p1y + p1z * p1z);
    p0f[0] = (float)(p0x / n0); p0f[1] = (float)(p0y / n0); p0f[2] = (float)(p0z / n0);
    p1f[0] = (float)(p1x / n1); p1f[1] = (float)(p1y / n1); p1f[2] = (float)(p1z / n1);
}

// Bc is row-major [12][48]. Cols 0..19: proj (rows 0..2 = P[v][axis]).
// Cols 20..39: sig2 quadratic form (rows 3..8 = [q0^2,q1^2,q2^2,2q0q1,2q0q2,2q1q2]).
__global__ void setup_B_kernel(float* __restrict__ Bc) {
    int c = threadIdx.x;
    if (c >= 48) return;
    float row[12];
#pragma unroll
    for (int k = 0; k < 12; ++k) row[k] = 0.0f;
    if (c < 40) {
        int cc = (c < 20) ? c : (c - 20);
        int v = cc >> 1, axis = cc & 1;
        float p0[3], p1[3];
        proj_axes(v, p0, p1);
        float q0 = axis ? p1[0] : p0[0];
        float q1 = axis ? p1[1] : p0[1];
        float q2 = axis ? p1[2] : p0[2];
        if (c < 20) {
            row[0] = q0; row[1] = q1; row[2] = q2;
        } else {
            row[3] = q0 * q0; row[4] = q1 * q1; row[5] = q2 * q2;
            row[6] = 2.0f * q0 * q1; row[7] = 2.0f * q0 * q2; row[8] = 2.0f * q1 * q2;
        }
    }
#pragma unroll
    for (int k = 0; k < 12; ++k) Bc[k * 48 + c] = row[k];
}

// ---------------- Projection GEMM via V_WMMA_F32_16X16X4_F32 ----------------
// pv layout: (N, 40): cols 0..19 = proj[v][axis], cols 20..39 = sig2[v][axis]

__device__ inline float get_in(const float* __restrict__ means, const float* __restrict__ covs,
                               int s, int k) {
    switch (k) {
        case 0: return means[s * 3 + 0];
        case 1: return means[s * 3 + 1];
        case 2: return means[s * 3 + 2];
        case 3: return covs[s * 9 + 0];  // C00
        case 4: return covs[s * 9 + 4];  // C11
        case 5: return covs[s * 9 + 8];  // C22
        case 6: return covs[s * 9 + 1];  // C01
        case 7: return covs[s * 9 + 2];  // C02
        case 8: return covs[s * 9 + 5];  // C12
        default: return 0.0f;
    }
}

__global__ void proj_wmma_kernel(const float* __restrict__ means, const float* __restrict__ covs,
                                 const float* __restrict__ Bc, float* __restrict__ pv) {
    int lane = threadIdx.x & 31;
    int wave = threadIdx.x >> 5;
    int tile = blockIdx.x * (blockDim.x >> 5) + wave;
    if (tile * 16 >= NS) return;  // wave-uniform
    int m = lane & 15;
    int h = lane >> 4;  // half-wave selects K pair
    int s = tile * 16 + m;

    // A (16x4 f32) per K-step: lanes0-15 hold (K=4k,4k+1), lanes16-31 hold (K=4k+2,4k+3)
    v2f a[3];
#pragma unroll
    for (int k = 0; k < 3; ++k) {
        a[k][0] = get_in(means, covs, s, 4 * k + 2 * h);
        a[k][1] = get_in(means, covs, s, 4 * k + 2 * h + 1);
    }

    v8f c0 = {}, c1 = {}, c2 = {};
#pragma unroll
    for (int k = 0; k < 3; ++k) {
        v2f b0, b1, b2;
        int r0 = (4 * k + 2 * h) * 48, r1 = (4 * k + 2 * h + 1) * 48;
        b0[0] = Bc[r0 + m];      b0[1] = Bc[r1 + m];
        b1[0] = Bc[r0 + 16 + m]; b1[1] = Bc[r1 + 16 + m];
        b2[0] = Bc[r0 + 32 + m]; b2[1] = Bc[r1 + 32 + m];
        c0 = __builtin_amdgcn_wmma_f32_16x16x4_f32(false, a[k], false, b0, (short)0, c0, false, false);
        c1 = __builtin_amdgcn_wmma_f32_16x16x4_f32(false, a[k], false, b1, (short)0, c1, false, false);
        c2 = __builtin_amdgcn_wmma_f32_16x16x4_f32(false, a[k], false, b2, (short)0, c2, false, false);
    }

    // D layout: VGPR r -> M = r (lanes 0-15) or r+8 (lanes 16-31); N = lane%16
#pragma unroll
    for (int r = 0; r < 8; ++r) {
        int sOut = tile * 16 + r + 8 * h;
        pv[(size_t)sOut * 40 + m] = c0[r];
        pv[(size_t)sOut * 40 + 16 + m] = c1[r];
        if (m < 8) pv[(size_t)sOut * 40 + 32 + m] = c2[r];
    }
}

// ---------------- Ripmap construction ----------------

__global__ void copy_base_kernel(const float4* __restrict__ fm4, float4* __restrict__ rip4) {
    int tid = blockIdx.x * blockDim.x + threadIdx.x;
    if (tid >= RES * RES * NV) return;
    int v = tid >> 18;           // / (512*512)
    int pix = tid & (RES * RES - 1);
    const float4* src = fm4 + ((size_t)v * RES * RES + pix) * 4;
    float4* dst = rip4 + ((size_t)v * T_PIX + pix) * 4;
#pragma unroll
    for (int q = 0; q < 4; ++q) dst[q] = src[q];
}

__global__ void pool_kernel(float* __restrict__ rip, int srcOff, int dstOff,
                            int Wd, int Hd, int Wsrc, int horiz) {
    int tid = blockIdx.x * blockDim.x + threadIdx.x;
    int planePix = Wd * Hd;
    if (tid >= planePix * NV) return;
    int v = tid / planePix;
    int pix = tid - v * planePix;
    int x = pix % Wd;
    int y = pix / Wd;
    size_t vbase = (size_t)v * T_PIX;
    size_t s0, s1;
    if (horiz) { s0 = (size_t)y * Wsrc + 2 * x; s1 = s0 + 1; }
    else       { s0 = (size_t)(2 * y) * Wsrc + x; s1 = s0 + (size_t)Wsrc; }
    const float4* pa = (const float4*)rip + (vbase + (size_t)srcOff + s0) * 4;
    const float4* pb = (const float4*)rip + (vbase + (size_t)srcOff + s1) * 4;
    float4* pd = (float4*)rip + (vbase + (size_t)dstOff + pix) * 4;
#pragma unroll
    for (int q = 0; q < 4; ++q) {
        float4 aq = pa[q], bq = pb[q], r;
        r.x = 0.5f * (aq.x + bq.x);
        r.y = 0.5f * (aq.y + bq.y);
        r.z = 0.5f * (aq.z + bq.z);
        r.w = 0.5f * (aq.w + bq.w);
        pd[q] = r;
    }
}

// ---------------- Quadrilinear gather ----------------

__device__ inline void acc_tap(float w, const float4* __restrict__ p,
                               float4& a0, float4& a1, float4& a2, float4& a3) {
    float4 t0 = p[0], t1 = p[1], t2 = p[2], t3 = p[3];
    a0.x = fmaf(w, t0.x, a0.x); a0.y = fmaf(w, t0.y, a0.y); a0.z = fmaf(w, t0.z, a0.z); a0.w = fmaf(w, t0.w, a0.w);
    a1.x = fmaf(w, t1.x, a1.x); a1.y = fmaf(w, t1.y, a1.y); a1.z = fmaf(w, t1.z, a1.z); a1.w = fmaf(w, t1.w, a1.w);
    a2.x = fmaf(w, t2.x, a2.x); a2.y = fmaf(w, t2.y, a2.y); a2.z = fmaf(w, t2.z, a2.z); a2.w = fmaf(w, t2.w, a2.w);
    a3.x = fmaf(w, t3.x, a3.x); a3.y = fmaf(w, t3.y, a3.y); a3.z = fmaf(w, t3.z, a3.z); a3.w = fmaf(w, t3.w, a3.w);
}

__global__ void gather_kernel(const float* __restrict__ pv, const float* __restrict__ rip,
                              float* __restrict__ out) {
    int n = blockIdx.x * blockDim.x + threadIdx.x;
    if (n >= NS) return;
    const float* pvn = pv + (size_t)n * 40;
    float* outn = out + (size_t)n * (NV * FEAT);
    for (int v = 0; v < NV; ++v) {
        float px = pvn[2 * v], py = pvn[2 * v + 1];
        float s2x = pvn[20 + 2 * v], s2y = pvn[21 + 2 * v];
        float lx = fminf(fmaxf(0.5f * log2f(fmaxf(s2x, 1e-20f)) + 9.0f, 0.0f), 7.0f);
        float ly = fminf(fmaxf(0.5f * log2f(fmaxf(s2y, 1e-20f)) + 9.0f, 0.0f), 7.0f);
        float l1ff = floorf(lx), l2ff = floorf(ly);
        float wx = lx - l1ff, wy = ly - l2ff;
        int l1a = (int)l1ff, l2a = (int)l2ff;
        int l1s[2] = {l1a, min(l1a + 1, 7)};
        int l2s[2] = {l2a, min(l2a + 1, 7)};
        float wl1[2] = {1.0f - wx, wx};
        float wl2[2] = {1.0f - wy, wy};

        float4 a0 = {0, 0, 0, 0}, a1 = {0, 0, 0, 0}, a2 = {0, 0, 0, 0}, a3 = {0, 0, 0, 0};
        const float4* ripv = (const float4*)rip + (size_t)v * T_PIX * 4;
#pragma unroll
        for (int i = 0; i < 2; ++i) {
            int l1 = l1s[i];
            int Wl = RES >> l1;
            float u = (px * 0.5f + 0.5f) * (float)Wl - 0.5f;
            float uf = floorf(u);
            float fu = u - uf;
            int ix0 = min(max((int)uf, 0), Wl - 1);
            int ix1 = min(max((int)uf + 1, 0), Wl - 1);
#pragma unroll
            for (int j = 0; j < 2; ++j) {
                int l2 = l2s[j];
                int Hl = RES >> l2;
                float vv = (py * 0.5f + 0.5f) * (float)Hl - 0.5f;
                float vf = floorf(vv);
                float fv = vv - vf;
                int iy0 = min(max((int)vf, 0), Hl - 1);
                int iy1 = min(max((int)vf + 1, 0), Hl - 1);
                int off = plane_off(l1, l2);
                float wl = wl1[i] * wl2[j];
                const float4* p00 = ripv + (size_t)(off + iy0 * Wl + ix0) * 4;
                const float4* p10 = ripv + (size_t)(off + iy0 * Wl + ix1) * 4;
                const float4* p01 = ripv + (size_t)(off + iy1 * Wl + ix0) * 4;
                const float4* p11 = ripv + (size_t)(off + iy1 * Wl + ix1) * 4;
                acc_tap(wl * (1.0f - fu) * (1.0f - fv), p00, a0, a1, a2, a3);
                acc_tap(wl * fu * (1.0f - fv),          p10, a0, a1, a2, a3);
                acc_tap(wl * (1.0f - fu) * fv,          p01, a0, a1, a2, a3);
                acc_tap(wl * fu * fv,                   p11, a0, a1, a2, a3);
            }
        }
        float4* op = (float4*)(outn + v * FEAT);
        op[0] = a0; op[1] = a1; op[2] = a2; op[3] = a3;
    }
}

// ---------------- Launch ----------------

extern "C" void kernel_launch(void* const* d_in, const int* in_sizes, int n_in,
                              void* d_out, int out_size, void* d_ws, size_t ws_size,
                              hipStream_t stream) {
    const float* means = (const float*)d_in[0];
    const float* covs = (const float*)d_in[1];
    const float* fm = (const float*)d_in[2];
    float* out = (float*)d_out;
    float* ws = (float*)d_ws;

    float* Bc = ws;                               // 12*48 floats
    float* pv = ws + 1024;                        // NS*40 floats
    float* rip = ws + 1024 + (size_t)NS * 40;     // NV*T_PIX*16 floats (~666 MB)

    setup_B_kernel<<<1, 64, 0, stream>>>(Bc);

    // projection GEMM: 16384 tiles of 16 samples, 8 waves/block
    proj_wmma_kernel<<<(NS / 16 + 7) / 8, 256, 0, stream>>>(means, covs, Bc, pv);

    // ripmap: base plane copy, then pooling cascade (dependency-ordered on stream)
    int basePix = RES * RES * NV;
    copy_base_kernel<<<(basePix + 255) / 256, 256, 0, stream>>>((const float4*)fm, (float4*)rip);
    for (int l1 = 1; l1 < 8; ++l1) {
        int Wd = RES >> l1, Hd = RES, Wsrc = RES >> (l1 - 1);
        int total = Wd * Hd * NV;
        pool_kernel<<<(total + 255) / 256, 256, 0, stream>>>(
            rip, plane_off(l1 - 1, 0), plane_off(l1, 0), Wd, Hd, Wsrc, 1);
    }
    for (int l1 = 0; l1 < 8; ++l1) {
        for (int l2 = 1; l2 < 8; ++l2) {
            int Wd = RES >> l1, Hd = RES >> l2;
            int total = Wd * Hd * NV;
            pool_kernel<<<(total + 255) / 256, 256, 0, stream>>>(
                rip, plane_off(l1, l2 - 1), plane_off(l1, l2), Wd, Hd, Wd, 0);
        }
    }

    gather_kernel<<<(NS + 255) / 256, 256, 0, stream>>>(pv, rip, out);
}